// MyGCNTopKPool_1194000908386
// MI455X (gfx1250) — compile-verified
//
#include <hip/hip_runtime.h>

typedef __attribute__((ext_vector_type(2))) float v2f;
typedef __attribute__((ext_vector_type(8))) float v8f;

#define Bg    256
#define Nn    400
#define Ee    6400
#define INdim 128
#define Hd    64
#define K1d   100
#define K2d   25
#define NTd   (Bg * Nn)
#define ETd   (Bg * Ee)
#define CHUNK 100

// ---------------- Dense GEMM: C[M x 64] = A[M x K] @ W[K x 64] (fp32 WMMA) ----
// Each wave computes a 16x64 tile with 4x V_WMMA_F32_16X16X4_F32 accumulators.
// A layout (16x4 f32): VGPR0 = {K=0 | lanes0-15, K=2 | lanes16-31}, VGPR1 = K=1/K=3.
// B layout (4x16): VGPR0 = rows K=0 / K=2 striped across lanes, VGPR1 = K=1/K=3.
// C/D: VGPR v -> M=v (lanes0-15), M=v+8 (lanes16-31), N = lane&15.
__global__ void __launch_bounds__(256) gemm64_wmma(const float* __restrict__ A,
                                                   const float* __restrict__ W,
                                                   float* __restrict__ C,
                                                   int M, int K) {
  int wave = (blockIdx.x * blockDim.x + threadIdx.x) >> 5;
  int lane = threadIdx.x & 31;
  int row0 = wave << 4;
  if (row0 >= M) return;  // wave-uniform: EXEC stays all-1s for WMMA
  int mrow = row0 + (lane & 15);
  int koff = (lane >> 4) << 1;  // 0 or 2
  const float* arow = A + (long)mrow * K;
  v8f acc[4] = {};
  for (int kb = 0; kb < K; kb += 4) {
    v2f a = *(const v2f*)(arow + kb + koff);
    int kr = kb + koff;
#pragma unroll
    for (int t = 0; t < 4; ++t) {
      int n = (t << 4) + (lane & 15);
      v2f b;
      b.x = W[kr * Hd + n];
      b.y = W[(kr + 1) * Hd + n];
      acc[t] = __builtin_amdgcn_wmma_f32_16x16x4_f32(false, a, false, b,
                                                     (short)0, acc[t],
                                                     false, false);
    }
  }
  int rbase = row0 + ((lane >> 4) << 3);
  int c0 = lane & 15;
#pragma unroll
  for (int t = 0; t < 4; ++t)
#pragma unroll
    for (int v = 0; v < 8; ++v)
      C[(long)(rbase + v) * Hd + (t << 4) + c0] = acc[t][v];
}

// ---------------- degree helpers ------------------------------------------
__global__ void set_ones_k(float* p, int n) {
  int i = blockIdx.x * blockDim.x + threadIdx.x;
  if (i < n) p[i] = 1.0f;
}
__global__ void deg_edges_k(const int* __restrict__ src, const int* __restrict__ dst,
                            float* __restrict__ deg, int ne) {
  int e = blockIdx.x * blockDim.x + threadIdx.x;
  if (e >= ne) return;
  int s = src[e], d = dst[e];
  if (s >= 0 && d >= 0) atomicAdd(&deg[d], 1.0f);
}
__global__ void rsqrt_k(const float* __restrict__ deg, float* __restrict__ dinv, int n) {
  int i = blockIdx.x * blockDim.x + threadIdx.x;
  if (i < n) dinv[i] = rsqrtf(deg[i]);
}

// ---------------- GCN aggregation (one workgroup per graph-chunk) ----------
// agg[d] += dinv[s]*dinv[d]*h[s]; out = relu(agg + h*dinv^2 + bias)
// One wave per edge: 32 lanes x 2 features, ds_add_f32 into LDS tile.
__global__ void __launch_bounds__(256) conv_agg_k(const float* __restrict__ h,
                                                  const float* __restrict__ dinv,
                                                  const float* __restrict__ bias,
                                                  const int* __restrict__ src,
                                                  const int* __restrict__ dst,
                                                  float* __restrict__ out,
                                                  int nper, int nchunks) {
  __shared__ float agg[CHUNK * Hd];
  int g = blockIdx.x / nchunks;
  int chunk = blockIdx.x % nchunks;
  int lo = chunk * CHUNK;
  int hi = lo + CHUNK; if (hi > nper) hi = nper;
  int nodes = hi - lo;
  for (int i = threadIdx.x; i < nodes * Hd; i += blockDim.x) agg[i] = 0.0f;
  __syncthreads();

  int lane = threadIdx.x & 31;
  int wave = threadIdx.x >> 5;
  int nw = blockDim.x >> 5;
  long ebase = (long)g * Ee;
  int gbase = g * nper + lo;
  for (int e = wave; e < Ee; e += nw) {
    int s = src[ebase + e];
    int d = dst[ebase + e];
    if (s < 0 || d < 0) continue;
    int dl = d - gbase;
    if ((unsigned)dl >= (unsigned)nodes) continue;
    float coef = dinv[s] * dinv[d];
    v2f hv = *(const v2f*)(h + (long)s * Hd + lane * 2);
    atomicAdd(&agg[dl * Hd + lane * 2],     coef * hv.x);
    atomicAdd(&agg[dl * Hd + lane * 2 + 1], coef * hv.y);
  }
  __syncthreads();

  for (int i = threadIdx.x; i < nodes * Hd; i += blockDim.x) {
    int nl = i >> 6, f = i & 63;
    long node = (long)gbase + nl;
    float hv = h[node * Hd + f];
    float di = dinv[node];
    float v = agg[i] + hv * di * di + bias[f];
    out[node * Hd + f] = v > 0.0f ? v : 0.0f;
  }
}

// ---------------- TopK pooling (one workgroup per graph) -------------------
// score = tanh(h.w/||w||); exact ranks reproduce lax.top_k stable ordering.
__global__ void __launch_bounds__(128) topk_pool_k(const float* __restrict__ hin,
                                                   const float* __restrict__ w,
                                                   float* __restrict__ xp,
                                                   int* __restrict__ nmap,
                                                   float* __restrict__ rsum,
                                                   int nper, int k) {
  __shared__ float sc[Nn];
  __shared__ float accf[Hd];
  __shared__ float invnorm_s;
  int g = blockIdx.x;
  if (threadIdx.x == 0) {
    float s = 0.0f;
    for (int f = 0; f < Hd; ++f) s += w[f] * w[f];
    invnorm_s = rsqrtf(s);
  }
  if (threadIdx.x < Hd) accf[threadIdx.x] = 0.0f;
  __syncthreads();
  float invnorm = invnorm_s;

  for (int n = threadIdx.x; n < nper; n += blockDim.x) {
    const float* row = hin + ((long)g * nper + n) * Hd;
    float d = 0.0f;
    for (int f = 0; f < Hd; ++f) d += row[f] * w[f];
    sc[n] = tanhf(d * invnorm);
  }
  __syncthreads();

  for (int n = threadIdx.x; n < nper; n += blockDim.x) {
    float sn = sc[n];
    int r = 0;
    for (int j = 0; j < nper; ++j) {
      float sj = sc[j];
      r += (sj > sn) || (sj == sn && j < n);
    }
    long gnode = (long)g * nper + n;
    if (r < k) {
      int newid = g * k + r;
      nmap[gnode] = newid;
      const float* row = hin + gnode * Hd;
      float* orow = xp + (long)newid * Hd;
      for (int f = 0; f < Hd; ++f) {
        float v = row[f] * sn;
        orow[f] = v;
        atomicAdd(&accf[f], v);  // per-graph pooled sum in LDS
      }
    } else {
      nmap[gnode] = -1;
    }
  }
  __syncthreads();
  if (threadIdx.x < Hd) rsum[g * Hd + threadIdx.x] = accf[threadIdx.x];
}

// ---------------- edge relabel after pooling -------------------------------
__global__ void remap_edges_k(const int* __restrict__ si, const int* __restrict__ di,
                              const int* __restrict__ nmap,
                              int* __restrict__ so, int* __restrict__ do_, int ne) {
  int e = blockIdx.x * blockDim.x + threadIdx.x;
  if (e >= ne) return;
  int s = si[e], d = di[e];
  int ns = -1, nd = -1;
  if (s >= 0 && d >= 0) { ns = nmap[s]; nd = nmap[d]; }
  if (ns < 0 || nd < 0) { ns = -1; nd = -1; }
  so[e] = ns; do_[e] = nd;
}

// ---------------- per-graph feature sum (for r3) ---------------------------
__global__ void __launch_bounds__(64) graph_sum_k(const float* __restrict__ xin,
                                                  float* __restrict__ r, int nper) {
  int g = blockIdx.x, f = threadIdx.x;
  float s = 0.0f;
  for (int n = 0; n < nper; ++n) s += xin[((long)g * nper + n) * Hd + f];
  r[g * Hd + f] = s;
}

// ---------------- final linear: out = [r3|r2|r1] @ Wl + bl -----------------
__global__ void final_linear_k(const float* __restrict__ r3, const float* __restrict__ r2,
                               const float* __restrict__ r1, const float* __restrict__ Wl,
                               const float* __restrict__ bl, float* __restrict__ out) {
  int idx = blockIdx.x * blockDim.x + threadIdx.x;
  if (idx >= Bg * 2) return;
  int g = idx >> 1, o = idx & 1;
  float s = bl[o];
  for (int f = 0; f < Hd; ++f) s += r3[g * Hd + f] * Wl[f * 2 + o];
  for (int f = 0; f < Hd; ++f) s += r2[g * Hd + f] * Wl[(Hd + f) * 2 + o];
  for (int f = 0; f < Hd; ++f) s += r1[g * Hd + f] * Wl[(2 * Hd + f) * 2 + o];
  out[idx] = s;
}

extern "C" void kernel_launch(void* const* d_in, const int* in_sizes, int n_in,
                              void* d_out, int out_size, void* d_ws, size_t ws_size,
                              hipStream_t stream) {
  const float* x   = (const float*)d_in[0];
  const int*   src = (const int*)d_in[1];
  const int*   dst = (const int*)d_in[2];
  const float* W1  = (const float*)d_in[4];
  const float* b1  = (const float*)d_in[5];
  const float* W2  = (const float*)d_in[6];
  const float* b2  = (const float*)d_in[7];
  const float* W3  = (const float*)d_in[8];
  const float* b3  = (const float*)d_in[9];
  const float* pw1 = (const float*)d_in[10];
  const float* pw2 = (const float*)d_in[11];
  const float* Wl  = (const float*)d_in[12];
  const float* bl  = (const float*)d_in[13];
  float* out = (float*)d_out;

  char* ws = (char*)d_ws;
  size_t off = 0;
  auto alloc = [&](size_t bytes) -> void* {
    off = (off + 255) & ~(size_t)255;
    void* p = ws + off;
    off += bytes;
    return p;
  };

  float* h1    = (float*)alloc((size_t)NTd * Hd * 4);
  float* c1    = (float*)alloc((size_t)NTd * Hd * 4);
  float* deg1  = (float*)alloc((size_t)NTd * 4);
  float* dinv1 = (float*)alloc((size_t)NTd * 4);
  int*   nmap1 = (int*)  alloc((size_t)NTd * 4);
  float* xp1   = (float*)alloc((size_t)Bg * K1d * Hd * 4);
  int*   es1   = (int*)  alloc((size_t)ETd * 4);
  int*   ed1   = (int*)  alloc((size_t)ETd * 4);
  float* h2    = (float*)alloc((size_t)Bg * K1d * Hd * 4);
  float* c2    = (float*)alloc((size_t)Bg * K1d * Hd * 4);
  float* deg2  = (float*)alloc((size_t)Bg * K1d * 4);
  float* dinv2 = (float*)alloc((size_t)Bg * K1d * 4);
  int*   nmap2 = (int*)  alloc((size_t)Bg * K1d * 4);
  float* xp2   = (float*)alloc((size_t)Bg * K2d * Hd * 4);
  int*   es2   = (int*)  alloc((size_t)ETd * 4);
  int*   ed2   = (int*)  alloc((size_t)ETd * 4);
  float* h3    = (float*)alloc((size_t)Bg * K2d * Hd * 4);
  float* c3    = (float*)alloc((size_t)Bg * K2d * Hd * 4);
  float* deg3  = (float*)alloc((size_t)Bg * K2d * 4);
  float* dinv3 = (float*)alloc((size_t)Bg * K2d * 4);
  float* r1    = (float*)alloc((size_t)Bg * Hd * 4);
  float* r2    = (float*)alloc((size_t)Bg * Hd * 4);
  float* r3    = (float*)alloc((size_t)Bg * Hd * 4);

  const int M1 = NTd, M2 = Bg * K1d, M3 = Bg * K2d;

  // ---- stage 1: GCNConv(IN->H) on full graph ----
  gemm64_wmma<<<M1 / 128, 256, 0, stream>>>(x, W1, h1, M1, INdim);
  set_ones_k<<<(M1 + 255) / 256, 256, 0, stream>>>(deg1, M1);
  deg_edges_k<<<(ETd + 255) / 256, 256, 0, stream>>>(src, dst, deg1, ETd);
  rsqrt_k<<<(M1 + 255) / 256, 256, 0, stream>>>(deg1, dinv1, M1);
  conv_agg_k<<<Bg * 4, 256, 0, stream>>>(h1, dinv1, b1, src, dst, c1, Nn, 4);

  // ---- pool 1: 400 -> 100 per graph, r1 = pooled sum ----
  topk_pool_k<<<Bg, 128, 0, stream>>>(c1, pw1, xp1, nmap1, r1, Nn, K1d);
  remap_edges_k<<<(ETd + 255) / 256, 256, 0, stream>>>(src, dst, nmap1, es1, ed1, ETd);

  // ---- stage 2: GCNConv(H->H) on pooled graph ----
  gemm64_wmma<<<M2 / 128, 256, 0, stream>>>(xp1, W2, h2, M2, Hd);
  set_ones_k<<<(M2 + 255) / 256, 256, 0, stream>>>(deg2, M2);
  deg_edges_k<<<(ETd + 255) / 256, 256, 0, stream>>>(es1, ed1, deg2, ETd);
  rsqrt_k<<<(M2 + 255) / 256, 256, 0, stream>>>(deg2, dinv2, M2);
  conv_agg_k<<<Bg, 256, 0, stream>>>(h2, dinv2, b2, es1, ed1, c2, K1d, 1);

  // ---- pool 2: 100 -> 25 per graph, r2 = pooled sum ----
  topk_pool_k<<<Bg, 128, 0, stream>>>(c2, pw2, xp2, nmap2, r2, K1d, K2d);
  remap_edges_k<<<(ETd + 255) / 256, 256, 0, stream>>>(es1, ed1, nmap2, es2, ed2, ETd);

  // ---- stage 3: GCNConv(H->H), r3 = per-graph sum ----
  gemm64_wmma<<<M3 / 128, 256, 0, stream>>>(xp2, W3, h3, M3, Hd);
  set_ones_k<<<(M3 + 255) / 256, 256, 0, stream>>>(deg3, M3);
  deg_edges_k<<<(ETd + 255) / 256, 256, 0, stream>>>(es2, ed2, deg3, ETd);
  rsqrt_k<<<(M3 + 255) / 256, 256, 0, stream>>>(deg3, dinv3, M3);
  conv_agg_k<<<Bg, 256, 0, stream>>>(h3, dinv3, b3, es2, ed2, c3, K2d, 1);
  graph_sum_k<<<Bg, 64, 0, stream>>>(c3, r3, K2d);

  // ---- readout: [B, 192] @ [192, 2] + bias ----
  final_linear_k<<<2, 256, 0, stream>>>(r3, r2, r1, Wl, bl, out);
}